// GATNet_84069689852490
// MI455X (gfx1250) — compile-verified
//
#include <hip/hip_runtime.h>
#include <hip/hip_bf16.h>

typedef float v2f __attribute__((ext_vector_type(2)));
typedef float v8f __attribute__((ext_vector_type(8)));

#define D 128
#define HD 64
#define ALPHA 0.2f

// ---------- float <-> ordered-uint for atomicMax on float ----------
__device__ __forceinline__ unsigned f2ord(float f) {
    unsigned u = __float_as_uint(f);
    return (u & 0x80000000u) ? ~u : (u | 0x80000000u);
}
__device__ __forceinline__ float ord2f(unsigned u) {
    unsigned v = (u & 0x80000000u) ? (u & 0x7FFFFFFFu) : ~u;
    return __uint_as_float(v);
}

// ---------- repack W[l,h,d,f] -> k-pair-major Wp[p][col][2], p = k/2 ----------
// Wp[(p*128 + col)*2 + q] = Wcat[k=2p+q][col], Wcat[k][h*64+f] = W[l,h,k,f].
// This makes each WMMA B operand (two consecutive K rows, same col) one b64 load.
__global__ void repack_w_kernel(const float* __restrict__ W, float* __restrict__ wp, int layer) {
    int idx = blockIdx.x * blockDim.x + threadIdx.x;
    if (idx >= D * D) return;
    int p   = idx >> 8;
    int col = (idx >> 1) & 127;
    int q   = idx & 1;
    int k   = 2 * p + q;
    int h   = col >> 6, f = col & 63;
    wp[idx] = W[(((size_t)(layer * 2 + h) * D) + k) * HD + f];
}

// ---------- GEMM: Hout[N,128] = X[N,128] @ W[128,128], fp32 WMMA ----------
// Wp staged in LDS once per 8-wave block; each wave computes a 16x128 strip:
// 8 accumulators x 32 k-steps of v_wmma_f32_16x16x4_f32, B fed by ds_load_b64.
__global__ __launch_bounds__(256) void gat_gemm_wmma(const float* __restrict__ X,
                                                     const float* __restrict__ Wp,
                                                     float* __restrict__ Hout, int N) {
    __shared__ float lw[D * D];              // 64 KB: pair-major weight tile
    {
        const float4* __restrict__ s4 = (const float4*)Wp;
        float4* d4 = (float4*)lw;
        #pragma unroll
        for (int i = 0; i < (D * D / 4) / 256; ++i)
            d4[i * 256 + threadIdx.x] = s4[i * 256 + threadIdx.x];
    }
    __syncthreads();

    int wave = blockIdx.x * (blockDim.x >> 5) + (threadIdx.x >> 5);
    int lane = threadIdx.x & 31;
    int row0 = wave * 16;
    if (row0 >= N) return;                   // wave-uniform: EXEC stays all-1s

    int r    = lane & 15;
    int koff = (lane < 16) ? 0 : 2;          // A layout: lanes 0-15 hold K=0,1; 16-31 K=2,3
    int koh  = koff >> 1;                    // pair offset
    int arow = row0 + r; if (arow >= N) arow = N - 1;
    const float* __restrict__ xrow = X + (size_t)arow * D;

    v8f acc[8] = {};
    #pragma unroll 4
    for (int k = 0; k < D; k += 4) {
        v2f a = *(const v2f*)(xrow + k + koff);              // b64 load
        const float* __restrict__ brow = lw + (((k >> 1) + koh) * D + r) * 2;
        #pragma unroll
        for (int c = 0; c < 8; ++c) {
            v2f b = *(const v2f*)(brow + c * 32);            // ds_load_b64
            acc[c] = __builtin_amdgcn_wmma_f32_16x16x4_f32(
                false, a, false, b, (short)0, acc[c], false, false);
        }
    }
    // C/D layout: lanes 0-15 -> M=j, lanes 16-31 -> M=8+j; N = r
    int mbase = row0 + ((lane < 16) ? 0 : 8);
    #pragma unroll
    for (int j = 0; j < 8; ++j) {
        int row = mbase + j;
        if (row >= N) continue;
        float* __restrict__ orow = Hout + (size_t)row * D;
        #pragma unroll
        for (int c = 0; c < 8; ++c) orow[c * 16 + r] = acc[c][j];
    }
}

// ---------- es/ed: per-(node,head) dot with attention vectors ----------
__global__ void attn_scores_kernel(const float* __restrict__ H,
                                   const float* __restrict__ a_src,
                                   const float* __restrict__ a_dst,
                                   float* __restrict__ es, float* __restrict__ ed,
                                   int N, int layer) {
    int idx = blockIdx.x * blockDim.x + threadIdx.x;
    if (idx >= 2 * N) return;
    int n = idx >> 1, h = idx & 1;
    const float* __restrict__ hv = H + (size_t)n * D + h * HD;
    const float* __restrict__ as = a_src + (size_t)(layer * 2 + h) * HD;
    const float* __restrict__ ad = a_dst + (size_t)(layer * 2 + h) * HD;
    float s = 0.f, d = 0.f;
    #pragma unroll 8
    for (int f = 0; f < HD; ++f) {
        float v = hv[f];
        s += v * as[f];
        d += v * ad[f];
    }
    es[(size_t)h * N + n] = s;
    ed[(size_t)h * N + n] = d;
}

// ---------- edge pass 1: leaky-relu logits + segment max (atomicMax) ----------
__global__ void edge_logits_kernel(const int* __restrict__ src, const int* __restrict__ dst,
                                   const float* __restrict__ es, const float* __restrict__ ed,
                                   float* __restrict__ ebuf, unsigned* __restrict__ menc,
                                   int E, int N) {
    int e = blockIdx.x * blockDim.x + threadIdx.x;
    if (e >= E) return;
    int s = src[e], d = dst[e];
    #pragma unroll
    for (int h = 0; h < 2; ++h) {
        float v = es[(size_t)h * N + s] + ed[(size_t)h * N + d];
        float l = (v > 0.f) ? v : ALPHA * v;
        ebuf[(size_t)e * 2 + h] = l;
        atomicMax(&menc[(size_t)d * 2 + h], f2ord(l));
    }
}

// ---------- edge pass 2: exp(e - max) + segment sum ----------
__global__ void edge_exp_kernel(const int* __restrict__ dst,
                                float* __restrict__ ebuf,
                                const unsigned* __restrict__ menc,
                                float* __restrict__ denom, int E) {
    int e = blockIdx.x * blockDim.x + threadIdx.x;
    if (e >= E) return;
    int d = dst[e];
    #pragma unroll
    for (int h = 0; h < 2; ++h) {
        float m = ord2f(menc[(size_t)d * 2 + h]);
        float x = __expf(ebuf[(size_t)e * 2 + h] - m);
        ebuf[(size_t)e * 2 + h] = x;
        atomicAdd(&denom[(size_t)d * 2 + h], x);
    }
}

// ---------- edge pass 3: acc[dst] += h[src] * att  (one wave per edge) ----------
__global__ __launch_bounds__(256) void edge_scatter_kernel(const int* __restrict__ src,
                                                           const int* __restrict__ dst,
                                                           const float* __restrict__ H,
                                                           const float* __restrict__ ebuf,
                                                           const float* __restrict__ denom,
                                                           float* __restrict__ acc, int E) {
    int wv = blockIdx.x * (blockDim.x >> 5) + (threadIdx.x >> 5);
    int lane = threadIdx.x & 31;
    if (wv >= E) return;
    int s = src[wv], d = dst[wv];
    int h = lane >> 4;                        // lanes 0-15: cols 0-63 (head0); 16-31: head1
    float att = ebuf[(size_t)wv * 2 + h] / (denom[(size_t)d * 2 + h] + 1e-16f);
    float4 v = ((const float4*)(H + (size_t)s * D))[lane];
    float* a = acc + (size_t)d * D + lane * 4;
    atomicAdd(a + 0, v.x * att);
    atomicAdd(a + 1, v.y * att);
    atomicAdd(a + 2, v.z * att);
    atomicAdd(a + 3, v.w * att);
}

// ---------- ELU in place ----------
__global__ void elu_kernel(float* __restrict__ x, size_t n) {
    size_t i = (size_t)blockIdx.x * blockDim.x + threadIdx.x;
    if (i >= n) return;
    float v = x[i];
    x[i] = (v > 0.f) ? v : (__expf(v) - 1.f);
}

// ---------- TransE: out[t] = G[h] + Rel[r] - G[t]  (float4 per lane) ----------
__global__ void transe_kernel(const float* __restrict__ G, const float* __restrict__ Rel,
                              const int* __restrict__ hl, const int* __restrict__ tl,
                              const int* __restrict__ rl, float* __restrict__ out, int T) {
    int tid = blockIdx.x * blockDim.x + threadIdx.x;
    if (tid >= T * 32) return;
    int t = tid >> 5, lane = tid & 31;
    float4 a = ((const float4*)(G   + (size_t)hl[t] * D))[lane];
    float4 b = ((const float4*)(Rel + (size_t)rl[t] * D))[lane];
    float4 c = ((const float4*)(G   + (size_t)tl[t] * D))[lane];
    float4 o;
    o.x = a.x + b.x - c.x;  o.y = a.y + b.y - c.y;
    o.z = a.z + b.z - c.z;  o.w = a.w + b.w - c.w;
    ((float4*)(out + (size_t)t * D))[lane] = o;
}

// ---------- seed gather ----------
__global__ void gather_kernel(const float* __restrict__ G, const int* __restrict__ idx,
                              float* __restrict__ out, int S) {
    int tid = blockIdx.x * blockDim.x + threadIdx.x;
    if (tid >= S * 32) return;
    int s = tid >> 5, lane = tid & 31;
    ((float4*)(out + (size_t)s * D))[lane] = ((const float4*)(G + (size_t)idx[s] * D))[lane];
}

extern "C" void kernel_launch(void* const* d_in, const int* in_sizes, int n_in,
                              void* d_out, int out_size, void* d_ws, size_t ws_size,
                              hipStream_t stream) {
    const float* ent_sr = (const float*)d_in[0];
    const float* ent_tg = (const float*)d_in[1];
    const float* rel_sr = (const float*)d_in[2];
    const float* rel_tg = (const float*)d_in[3];
    const float* W      = (const float*)d_in[4];
    const float* a_src  = (const float*)d_in[5];
    const float* a_dst  = (const float*)d_in[6];
    const int* edge_sr  = (const int*)d_in[7];
    const int* edge_tg  = (const int*)d_in[8];
    const int* sr_data  = (const int*)d_in[9];
    const int* tg_data  = (const int*)d_in[10];
    const int* h_sr = (const int*)d_in[11];
    const int* h_tg = (const int*)d_in[12];
    const int* t_sr = (const int*)d_in[13];
    const int* t_tg = (const int*)d_in[14];
    const int* r_sr = (const int*)d_in[15];
    const int* r_tg = (const int*)d_in[16];

    const int Nsr = in_sizes[0] / D;
    const int Ntg = in_sizes[1] / D;
    const int Esr = in_sizes[7] / 2;
    const int Etg = in_sizes[8] / 2;
    const int S   = in_sizes[9];
    const int T   = in_sizes[11];
    const int Nmax = Nsr > Ntg ? Nsr : Ntg;
    const int Emax = Esr > Etg ? Esr : Etg;

    // ---- carve workspace ----
    char* p = (char*)d_ws;
    auto carve = [&](size_t bytes) {
        void* q = (void*)p;
        p += (bytes + 255) & ~(size_t)255;
        return q;
    };
    float*    bufA  = (float*)carve((size_t)Nmax * D * 4);
    float*    bufB  = (float*)carve((size_t)Nmax * D * 4);
    float*    bufC  = (float*)carve((size_t)Nmax * D * 4);
    float*    wp0   = (float*)carve((size_t)D * D * 4);
    float*    wp1   = (float*)carve((size_t)D * D * 4);
    float*    es    = (float*)carve((size_t)2 * Nmax * 4);
    float*    edv   = (float*)carve((size_t)2 * Nmax * 4);
    float*    denom = (float*)carve((size_t)2 * Nmax * 4);
    unsigned* menc  = (unsigned*)carve((size_t)2 * Nmax * 4);
    float*    ebuf  = (float*)carve((size_t)2 * Emax * 4);

    float* out         = (float*)d_out;
    float* out_seed_sr = out;
    float* out_seed_tg = out + (size_t)S * D;
    float* out_tr_sr   = out + (size_t)2 * S * D;
    float* out_tr_tg   = out_tr_sr + (size_t)T * D;

    // ---- repack W for both layers (shared across KGs) ----
    repack_w_kernel<<<(D * D + 255) / 256, 256, 0, stream>>>(W, wp0, 0);
    repack_w_kernel<<<(D * D + 255) / 256, 256, 0, stream>>>(W, wp1, 1);

    // ---- full pipeline for one KG ----
    auto run_kg = [&](const float* ent, const int* edges, int N, int E,
                      const float* rel, const int* hl, const int* tl, const int* rl,
                      const int* seed, float* seed_out, float* tr_out) {
        const int* srcl = edges;        // edge row 0
        const int* dstl = edges + E;    // edge row 1
        const size_t NDbytes = (size_t)N * D * 4;

        hipMemcpyAsync(bufA, ent, NDbytes, hipMemcpyDeviceToDevice, stream);
        float* x   = bufA;
        float* hb  = bufB;
        float* acc = bufC;

        const int gemm_waves  = (N + 15) / 16;
        const int gemm_blocks = (gemm_waves + 7) / 8;       // 256 thr = 8 waves
        const int eth_blocks  = (E + 255) / 256;
        const int sc_blocks   = (E + 7) / 8;                // wave per edge
        const int nh_blocks   = (2 * N + 255) / 256;

        for (int l = 0; l < 2; ++l) {
            const float* wc = (l == 0) ? wp0 : wp1;
            gat_gemm_wmma<<<gemm_blocks, 256, 0, stream>>>(x, wc, hb, N);
            attn_scores_kernel<<<nh_blocks, 256, 0, stream>>>(hb, a_src, a_dst, es, edv, N, l);
            hipMemsetAsync(menc,  0, (size_t)2 * N * 4, stream);
            hipMemsetAsync(denom, 0, (size_t)2 * N * 4, stream);
            hipMemsetAsync(acc,   0, NDbytes, stream);
            edge_logits_kernel<<<eth_blocks, 256, 0, stream>>>(srcl, dstl, es, edv, ebuf, menc, E, N);
            edge_exp_kernel<<<eth_blocks, 256, 0, stream>>>(dstl, ebuf, menc, denom, E);
            edge_scatter_kernel<<<sc_blocks, 256, 0, stream>>>(srcl, dstl, hb, ebuf, denom, acc, E);
            if (l == 0) {
                elu_kernel<<<(int)(((size_t)N * D + 255) / 256), 256, 0, stream>>>(acc, (size_t)N * D);
                float* oldx = x;
                x = acc;        // ELU'd output becomes next layer input
                acc = oldx;     // recycle old input as next accumulator
            }
        }
        const float* G = acc;   // final layer output (acc of layer 1)
        gather_kernel<<<(S * 32 + 255) / 256, 256, 0, stream>>>(G, seed, seed_out, S);
        transe_kernel<<<(T * 32 + 255) / 256, 256, 0, stream>>>(G, rel, hl, tl, rl, tr_out, T);
    };

    run_kg(ent_sr, edge_sr, Nsr, Esr, rel_sr, h_sr, t_sr, r_sr, sr_data, out_seed_sr, out_tr_sr);
    run_kg(ent_tg, edge_tg, Ntg, Etg, rel_tg, h_tg, t_tg, r_tg, tg_data, out_seed_tg, out_tr_tg);

    (void)n_in; (void)out_size; (void)ws_size;
}